// EMAEnergyNormalizer_53953379173289
// MI455X (gfx1250) — compile-verified
//
#include <hip/hip_runtime.h>

#define B_ 32
#define F_ 257
#define T_ 3000
#define FT_ (F_ * T_)            // 771000
#define MOM 0.99f
#define OMM 0.01f
#define EPS_ 1e-8f

typedef __attribute__((ext_vector_type(2))) float v2f;
typedef __attribute__((ext_vector_type(8))) float v8f;

// ---------------------------------------------------------------------------
// Kernel A: frame means over F via V_WMMA_F32_16X16X4_F32 (ones-vector trick).
// One wave handles a 16-wide t-tile for one batch. A = ones(16x4), B holds a
// 4(f) x 16(t) slab of mag; D row 0 = column sums. 64 WMMAs cover f=0..255,
// row f=256 added on VALU.
//
// Out-of-range columns use a CLAMPED address (no per-iteration exec masking:
// a D column only mixes its own lane's B data, so garbage sums in tail lanes
// are harmless — the store guard drops them). The loop is software-pipelined
// so next-chunk loads overlap the current WMMA.
// ---------------------------------------------------------------------------
__global__ __launch_bounds__(256) void fmean_kernel(const float* __restrict__ mag,
                                                    float* __restrict__ fmean) {
    const int lane = threadIdx.x & 31;
    const int wave = threadIdx.x >> 5;
    const int b    = blockIdx.y;
    const int col  = blockIdx.x * 128 + wave * 16 + (lane & 15);
    const int colc = (col < T_) ? col : (T_ - 1);   // clamp: no divergence in loop
    const int half = lane >> 4;   // 0: lanes 0-15 (K even rows), 1: lanes 16-31

    const float* base = mag + (size_t)b * FT_ + colc;

#if __has_builtin(__builtin_amdgcn_wmma_f32_16x16x4_f32)
    v8f c = {};
    v2f a; a[0] = 1.0f; a[1] = 1.0f;

    // prologue: load chunk fc=0
    v2f bcur;
    bcur[0] = base[(size_t)(0 + half) * T_];        // B VGPR0: K=0 (lo lanes) / K=1 (hi lanes)
    bcur[1] = base[(size_t)(2 + half) * T_];        // B VGPR1: K=2 / K=3

#pragma unroll 4
    for (int fc = 0; fc < 252; fc += 4) {
        v2f bnext;
        bnext[0] = base[(size_t)(fc + 4 + half) * T_];
        bnext[1] = base[(size_t)(fc + 6 + half) * T_];
        c = __builtin_amdgcn_wmma_f32_16x16x4_f32(false, a, false, bcur,
                                                  (short)0, c, false, false);
        bcur = bnext;
    }
    c = __builtin_amdgcn_wmma_f32_16x16x4_f32(false, a, false, bcur,
                                              (short)0, c, false, false);
    float sum = c[0];   // D[M=0 (or 8), N=lane&15]: identical sums in every row
#else
    float sum = 0.0f;
    for (int f = 0; f < 256; ++f) sum += base[(size_t)f * T_];
#endif
    if ((lane < 16) && (col < T_)) {
        sum += base[(size_t)256 * T_];               // leftover row f=256
        fmean[(size_t)b * T_ + col] = sum * (1.0f / 257.0f);
    }
}

// ---------------------------------------------------------------------------
// Kernel B: decoupled-chunk EMA scan. 8 chunks x 32 batches, one block.
// m_end(chunk) = 0.99^CH * m_start + s  (linearity), carries combined in LDS.
// Emits mean_with_bias into the mag_mean output slab and 1/(val+eps) into ws.
// ---------------------------------------------------------------------------
__global__ __launch_bounds__(256) void ema_kernel(const float* __restrict__ fmean,
                                                  const float* __restrict__ bias,
                                                  const float* __restrict__ rmean,
                                                  float* __restrict__ out_mean,
                                                  float* __restrict__ inv) {
    const int b = threadIdx.x & 31;   // batch
    const int c = threadIdx.x >> 5;   // chunk 0..7
    const int CH = T_ / 8;            // 375

    __shared__ float s_sh[8][32];
    __shared__ float p_sh[8][32];
    __shared__ float carry_sh[8][32];

    const float* fp = fmean + (size_t)b * T_ + c * CH;

    // Pass 1: local scan with zero init + decay factor
    float s = 0.0f, pw = 1.0f;
#pragma unroll 5
    for (int i = 0; i < CH; ++i) {
        s  = MOM * s + OMM * fp[i];
        pw *= MOM;
    }
    s_sh[c][b] = s;
    p_sh[c][b] = pw;
    __syncthreads();

    // Serial carry combine (7 steps) — one thread per batch
    if (c == 0) {
        float m = rmean[0];
        carry_sh[0][b] = m;
        for (int j = 0; j < 7; ++j) {
            m = p_sh[j][b] * m + s_sh[j][b];
            carry_sh[j + 1][b] = m;
        }
    }
    __syncthreads();

    // Pass 2: re-scan with true carry, emit outputs
    float m = carry_sh[c][b];
    const float bias0 = bias[0];
    float* om = out_mean + (size_t)b * T_ + c * CH;
    float* iv = inv      + (size_t)b * T_ + c * CH;
#pragma unroll 5
    for (int i = 0; i < CH; ++i) {
        m = MOM * m + OMM * fp[i];
        float val = m + bias0;
        om[i] = val;
        iv[i] = 1.0f / (val + EPS_);
    }
}

// ---------------------------------------------------------------------------
// Kernel C: mag_norm = mag * inv[b,t]. Streaming b128 loads/stores, 4 el/thread.
// A float4 never crosses a batch (FT_ % 4 == 0); handle t-wrap per element.
// ---------------------------------------------------------------------------
__global__ __launch_bounds__(256) void norm_kernel(const float* __restrict__ mag,
                                                   const float* __restrict__ inv,
                                                   float* __restrict__ out) {
    const int FT4 = FT_ / 4;                       // 192750
    const int n4  = B_ * FT4;                      // 6,168,000
    int gid = blockIdx.x * blockDim.x + threadIdx.x;
    if (gid >= n4) return;

    int b    = gid / FT4;
    int rem  = (gid - b * FT4) * 4;
    int t    = rem % T_;

    float4 v = reinterpret_cast<const float4*>(mag)[gid];
    const float* ivb = inv + (size_t)b * T_;

    int t1 = t + 1; if (t1 >= T_) t1 -= T_;
    int t2 = t + 2; if (t2 >= T_) t2 -= T_;
    int t3 = t + 3; if (t3 >= T_) t3 -= T_;

    v.x *= ivb[t];
    v.y *= ivb[t1];
    v.z *= ivb[t2];
    v.w *= ivb[t3];

    reinterpret_cast<float4*>(out)[gid] = v;
}

extern "C" void kernel_launch(void* const* d_in, const int* in_sizes, int n_in,
                              void* d_out, int out_size, void* d_ws, size_t ws_size,
                              hipStream_t stream) {
    const float* mag   = (const float*)d_in[0];
    const float* bias  = (const float*)d_in[1];
    const float* rmean = (const float*)d_in[2];
    float* out   = (float*)d_out;
    float* fmean = (float*)d_ws;                   // B*T floats
    float* inv   = fmean + (size_t)B_ * T_;        // B*T floats

    // A: per-frame means (WMMA reduction over F)
    dim3 gridA((T_ + 127) / 128, B_);
    fmean_kernel<<<gridA, 256, 0, stream>>>(mag, fmean);

    // B: EMA scan -> mag_mean slab of d_out + reciprocal table
    ema_kernel<<<1, 256, 0, stream>>>(fmean, bias, rmean,
                                      out + (size_t)B_ * FT_, inv);

    // C: streaming normalize
    const int n4 = B_ * (FT_ / 4);
    norm_kernel<<<(n4 + 255) / 256, 256, 0, stream>>>(mag, inv, out);
}